// RadialBasis_61976378081998
// MI455X (gfx1250) — compile-verified
//
#include <hip/hip_runtime.h>

// RadialBasis: out[e, l, n] = j_l( zeros[n,l] * r[e] / CUTOFF ),  out: [N, 8, 10] f32.
// Roofline: 640MB NT stores @ 23.3TB/s ~= 28us floor; ~150 hw transcendentals/edge
// (~19 cyc/edge at 1/4-rate trans) is the matching compute bound -> kernel sits at
// the roofline crossover. No contraction structure -> WMMA inapplicable.
// CDNA5 paths: global_load_async_to_lds_b32 + s_wait_asynccnt staging,
// global_prefetch_b8, packed VOP3P f32 pair math, non-temporal stores.
//
// Mapping: 1 thread = 1 edge; l is compile-time per row (exact-length recurrence,
// no predication waste); n-values processed as float2 pairs for packed-f32 ALUs.

#define NMAX    10
#define LMAXP1  8
#define NL      80
#define INV_CUTOFF 0.2f

typedef float v2f __attribute__((ext_vector_type(2)));

#if __has_builtin(__builtin_amdgcn_global_load_async_to_lds_b32) && \
    __has_builtin(__builtin_amdgcn_s_wait_asynccnt)
#define USE_ASYNC_LDS 1
typedef __attribute__((address_space(1))) int* gas_i32p;
typedef __attribute__((address_space(3))) int* las_i32p;
#else
#define USE_ASYNC_LDS 0
#endif

// Raw v_rcp_f32 (~1 ulp): avoids the 10-op IEEE divide chain on the per-edge
// critical path; downstream values are approximation products anyway.
__device__ __forceinline__ float fast_rcp(float x) {
#if __has_builtin(__builtin_amdgcn_rcpf)
    return __builtin_amdgcn_rcpf(x);
#else
    return 1.0f / x;
#endif
}

// One l-row: 10 values as 5 packed pairs. L is compile time -> recurrence has
// exactly max(L-1,0) steps; pair arithmetic lowers to v_pk_{mul,fma,add}_f32.
template <int L>
__device__ __forceinline__ void bessel_row(const float* __restrict__ sZT,
                                           const float* __restrict__ sIT,
                                           float t, float invt,
                                           float* __restrict__ o)
{
#pragma unroll
    for (int p = 0; p < NMAX / 2; ++p) {
        const v2f z   = *(const v2f*)&sZT[L * NMAX + 2 * p];  // ds_load_2addr (broadcast)
        const v2f iz  = *(const v2f*)&sIT[L * NMAX + 2 * p];
        const v2f x   = z * t;
        const v2f inv = iz * invt;                            // 1/x, no divide
        if (L == 0) {
            // j_0 = sin(x)/x : no cos needed for this row
            v2f s; s.x = __sinf(x.x); s.y = __sinf(x.y);
            __builtin_nontemporal_store(s * inv, (v2f*)o + p);
        } else {
            float s0, c0, s1, c1;
            __sincosf(x.x, &s0, &c0);
            __sincosf(x.y, &s1, &c1);
            const v2f s = { s0, s1 }, c = { c0, c1 };
            const v2f j0 = s * inv;
            v2f jm1 = j0;
            v2f jc  = (j0 - c) * inv;                         // j_1
#pragma unroll
            for (int k = 2; k <= L; ++k) {                    // j_k=(2k-1)/x*j_{k-1}-j_{k-2}
                const v2f jn = ((2.0f * k - 1.0f) * inv) * jc - jm1;  // pk_mul + pk_fma
                jm1 = jc; jc = jn;
            }
            __builtin_nontemporal_store(jc, (v2f*)o + p);
        }
    }
}

__global__ __launch_bounds__(256) void RadialBasis_61976378081998_kernel(
    const float* __restrict__ r,
    const float* __restrict__ zeros,   // [NMAX, LMAX+1]: zeros[n*8 + l]
    float* __restrict__ out,           // [N, 8, 10]
    int n_edges)
{
    __shared__ float sRaw[NL];         // as stored in memory: [n][l]
    __shared__ float sZT[NL];          // transposed: [l][n]
    __shared__ float sIT[NL];          // 1/z transposed: [l][n]
    const int tid = threadIdx.x;

    // ---- Stage zeros into LDS via async DMA path, then transpose + rcp ----
#if USE_ASYNC_LDS
    if (tid < NL) {
        __builtin_amdgcn_global_load_async_to_lds_b32(
            (gas_i32p)(zeros + tid), (las_i32p)(&sRaw[tid]), 0, 0);
    }
    __builtin_amdgcn_s_wait_asynccnt(0);
    __syncthreads();
#else
    if (tid < NL) sRaw[tid] = zeros[tid];
    __syncthreads();
#endif
    if (tid < NL) {
        const int n = tid >> 3;        // tid / 8
        const int l = tid & 7;         // tid % 8
        const float z = sRaw[tid];
        sZT[l * NMAX + n] = z;
        sIT[l * NMAX + n] = fast_rcp(z);   // v_rcp_f32, once per block
    }
    __syncthreads();

    const int e = blockIdx.x * 256 + tid;
    if (e >= n_edges) return;

    // Stream r[] ahead of the trans-heavy body (global_prefetch_b8).
    {
        const int pe = e + 16384;
        if (pe < n_edges) __builtin_prefetch(r + pe, 0, 1);
    }

    const float rv   = r[e];
    const float t    = rv * INV_CUTOFF;
    const float invt = fast_rcp(t);                  // single v_rcp_f32 per edge

    float* o = out + (size_t)e * NL;                 // 320 contiguous bytes per edge
    bessel_row<0>(sZT, sIT, t, invt, o + 0 * NMAX);
    bessel_row<1>(sZT, sIT, t, invt, o + 1 * NMAX);
    bessel_row<2>(sZT, sIT, t, invt, o + 2 * NMAX);
    bessel_row<3>(sZT, sIT, t, invt, o + 3 * NMAX);
    bessel_row<4>(sZT, sIT, t, invt, o + 4 * NMAX);
    bessel_row<5>(sZT, sIT, t, invt, o + 5 * NMAX);
    bessel_row<6>(sZT, sIT, t, invt, o + 6 * NMAX);
    bessel_row<7>(sZT, sIT, t, invt, o + 7 * NMAX);
}

extern "C" void kernel_launch(void* const* d_in, const int* in_sizes, int n_in,
                              void* d_out, int out_size, void* d_ws, size_t ws_size,
                              hipStream_t stream) {
    const float* r     = (const float*)d_in[0];
    const float* zeros = (const float*)d_in[1];
    float*       out   = (float*)d_out;
    const int n_edges  = in_sizes[0];

    const int blocks = (n_edges + 255) / 256;
    RadialBasis_61976378081998_kernel<<<blocks, 256, 0, stream>>>(r, zeros, out, n_edges);
}